// GraphQNetwork_84524956385327
// MI455X (gfx1250) — compile-verified
//
#include <hip/hip_runtime.h>
#include <math.h>

#define NN    50000
#define EE    1600000
#define GG    64
#define FNODE 128
#define FEDGE 16
#define ACTD  10
#define NEG_SLOPE 0.2f

typedef float v2f __attribute__((ext_vector_type(2)));
typedef float v8f __attribute__((ext_vector_type(8)));

// Native FP32 WMMA: D(16x16,f32) = A(16x4,f32) x B(4x16,f32) + C
// A: lanes 0-15 -> M=0..15 holding K={0,1}; lanes 16-31 -> same M, K={2,3}
// B: lanes 0-15 -> N=0..15 holding K={0,1}; lanes 16-31 -> same N, K={2,3}
// C/D: VGPR j: lanes 0-15 -> M=j; lanes 16-31 -> M=j+8; N = lane&15
__device__ __forceinline__ v8f wmma4(v2f a, v2f b, v8f c) {
  return __builtin_amdgcn_wmma_f32_16x16x4_f32(false, a, false, b, (short)0, c,
                                               false, false);
}

// ---------------------------------------------------------------- utilities
__global__ void fill_f32(float* __restrict__ p, float v, int n) {
  int i = blockIdx.x * blockDim.x + threadIdx.x;
  if (i < n) p[i] = v;
}

// ------------------------------------------------- dual GEMM: xl & xr (WMMA)
// XL = H @ Wl + bl ; XR = H @ Wr + br.  H:[NN,din] Wl/Wr:[din,dout]
// One wave per 16x16 output tile; shared A operand for both products.
__global__ void gemm_xlxr_wmma(const float* __restrict__ H,
                               const float* __restrict__ Wl, const float* __restrict__ bl,
                               const float* __restrict__ Wr, const float* __restrict__ br,
                               float* __restrict__ XL, float* __restrict__ XR,
                               int din, int dout, int ntilesN, int totalTiles) {
  int wave = blockIdx.x * (blockDim.x >> 5) + (threadIdx.x >> 5);
  if (wave >= totalTiles) return;            // wave-uniform exit (EXEC all-1 for WMMA)
  int mt = wave / ntilesN;
  int nt = wave % ntilesN;

  int lane = threadIdx.x & 31;
  int l15  = lane & 15;
  int kb   = (lane >> 4) << 1;               // 0 (lanes 0-15) or 2 (lanes 16-31)
  int row  = mt * 16 + l15;
  int col  = nt * 16 + l15;

  const float* hrow = H + (size_t)row * din;
  v8f cL = {};
  v8f cR = {};
  for (int k = 0; k < din; k += 4) {
    v2f a;  a.x  = hrow[k + kb];                    a.y  = hrow[k + kb + 1];
    v2f b0; b0.x = Wl[(size_t)(k + kb) * dout + col]; b0.y = Wl[(size_t)(k + kb + 1) * dout + col];
    v2f b1; b1.x = Wr[(size_t)(k + kb) * dout + col]; b1.y = Wr[(size_t)(k + kb + 1) * dout + col];
    cL = wmma4(a, b0, cL);
    cR = wmma4(a, b1, cR);
  }
  float blc = bl[col], brc = br[col];
  int rbase = mt * 16 + ((lane >> 4) << 3);  // +0 or +8
  #pragma unroll
  for (int j = 0; j < 8; ++j) {
    int r = rbase + j;
    XL[(size_t)r * dout + col] = cL[j] + blc;
    XR[(size_t)r * dout + col] = cR[j] + brc;
  }
}

// ------------------------------------------ pass A: per-edge score + seg-max
__global__ void edge_score(const float* __restrict__ XL, const float* __restrict__ XR,
                           const float* __restrict__ EA,
                           const int* __restrict__ srcI, const int* __restrict__ dstI,
                           const float* __restrict__ We, const float* __restrict__ att,
                           float* __restrict__ score, float* __restrict__ mx, int dout) {
  extern __shared__ float lds[];             // [16*dout] We + [dout] att
  float* sWe  = lds;
  float* sAtt = lds + FEDGE * dout;
  for (int i = threadIdx.x; i < FEDGE * dout; i += blockDim.x) sWe[i] = We[i];
  for (int i = threadIdx.x; i < dout;        i += blockDim.x) sAtt[i] = att[i];
  __syncthreads();

  int e = blockIdx.x * blockDim.x + threadIdx.x;
  if (e >= EE) return;
  int s = srcI[e], d = dstI[e];

  const float* ep = EA + (size_t)e * FEDGE;
  float ea[FEDGE];
  #pragma unroll
  for (int k = 0; k < FEDGE; ++k) ea[k] = ep[k];

  const float* xlp = XL + (size_t)s * dout;
  const float* xrp = XR + (size_t)d * dout;
  __builtin_prefetch(xlp, 0, 0);
  __builtin_prefetch(xrp, 0, 0);

  float sc = 0.0f;
  for (int f = 0; f < dout; ++f) {
    float ew = 0.0f;
    #pragma unroll
    for (int k = 0; k < FEDGE; ++k) ew = fmaf(ea[k], sWe[k * dout + f], ew);
    float m  = xlp[f] + xrp[f] + ew;
    float lr = (m > 0.0f) ? m : NEG_SLOPE * m;
    sc = fmaf(lr, sAtt[f], sc);
  }
  score[e] = sc;
  // float atomic max via signed/unsigned trick (mx initialized to -inf)
  if (sc >= 0.0f) atomicMax((int*)(mx + d), __float_as_int(sc));
  else            atomicMin((unsigned int*)(mx + d), __float_as_uint(sc));
}

// ---------------------------------------------- pass B: exp + seg-sum denom
__global__ void edge_exp(const float* __restrict__ score, const float* __restrict__ mx,
                         const int* __restrict__ dstI,
                         float* __restrict__ ex, float* __restrict__ den) {
  int e = blockIdx.x * blockDim.x + threadIdx.x;
  if (e >= EE) return;
  int d = dstI[e];
  float v = expf(score[e] - mx[d]);
  ex[e] = v;
  atomicAdd(den + d, v);
}

// --------------------------------------- pass C: weighted scatter-aggregate
__global__ void edge_agg(const float* __restrict__ XL,
                         const float* __restrict__ ex, const float* __restrict__ den,
                         const int* __restrict__ srcI, const int* __restrict__ dstI,
                         float* __restrict__ agg, int dout) {
  int e = blockIdx.x * blockDim.x + threadIdx.x;
  if (e >= EE) return;
  int s = srcI[e], d = dstI[e];
  float alpha = ex[e] / (den[d] + 1e-16f);
  const float* xlp = XL + (size_t)s * dout;
  float*       op  = agg + (size_t)d * dout;
  __builtin_prefetch(xlp, 0, 0);
  for (int f = 0; f < dout; ++f) atomicAdd(op + f, alpha * xlp[f]);
}

// ------------------------------------------------------- bias + ReLU (node)
__global__ void node_bias_relu(float* __restrict__ h, const float* __restrict__ bo,
                               int dout, int total) {
  int i = blockIdx.x * blockDim.x + threadIdx.x;
  if (i >= total) return;
  float v = h[i] + bo[i % dout];
  h[i] = v > 0.0f ? v : 0.0f;
}

// ------------------------------------------------------------- mean pooling
__global__ void pool_count(const int* __restrict__ batch, float* __restrict__ cnt) {
  int n = blockIdx.x * blockDim.x + threadIdx.x;
  if (n < NN) atomicAdd(cnt + batch[n], 1.0f);
}
__global__ void pool_sum(const float* __restrict__ h, const int* __restrict__ batch,
                         float* __restrict__ pooled) {
  int i = blockIdx.x * blockDim.x + threadIdx.x;
  if (i >= NN * FNODE) return;
  int n = i / FNODE, f = i % FNODE;
  atomicAdd(pooled + (size_t)batch[n] * FNODE + f, h[i]);
}

// ---------------------------------------------------- dueling heads (tiny)
__global__ void heads(const float* __restrict__ pooled, const float* __restrict__ cnt,
                      const float* __restrict__ Wv1, const float* __restrict__ bv1,
                      const float* __restrict__ Wv2, const float* __restrict__ bv2,
                      const float* __restrict__ Wa1, const float* __restrict__ ba1,
                      const float* __restrict__ Wa2, const float* __restrict__ ba2,
                      float* __restrict__ out) {
  __shared__ float pg[FNODE];
  __shared__ float hv[64], ha[64], aa[ACTD], sv, amean;
  int g = blockIdx.x, t = threadIdx.x;
  float ic = 1.0f / fmaxf(cnt[g], 1.0f);
  if (t < FNODE) pg[t] = pooled[(size_t)g * FNODE + t] * ic;
  __syncthreads();
  if (t < 64) {
    float av = bv1[t], aw = ba1[t];
    for (int k = 0; k < FNODE; ++k) {
      av = fmaf(pg[k], Wv1[k * 64 + t], av);
      aw = fmaf(pg[k], Wa1[k * 64 + t], aw);
    }
    hv[t] = fmaxf(av, 0.0f);
    ha[t] = fmaxf(aw, 0.0f);
  }
  __syncthreads();
  if (t == 0) {
    float v = bv2[0];
    for (int k = 0; k < 64; ++k) v = fmaf(hv[k], Wv2[k], v);
    sv = v;
  }
  if (t < ACTD) {
    float a = ba2[t];
    for (int k = 0; k < 64; ++k) a = fmaf(ha[k], Wa2[k * ACTD + t], a);
    aa[t] = a;
  }
  __syncthreads();
  if (t == 0) {
    float m = 0.0f;
    for (int j = 0; j < ACTD; ++j) m += aa[j];
    amean = m / (float)ACTD;
  }
  __syncthreads();
  if (t < ACTD) out[g * ACTD + t] = sv + aa[t] - amean;
}

// ============================================================== launcher
extern "C" void kernel_launch(void* const* d_in, const int* in_sizes, int n_in,
                              void* d_out, int out_size, void* d_ws, size_t ws_size,
                              hipStream_t stream) {
  (void)in_sizes; (void)n_in; (void)out_size; (void)ws_size;
  const float* x     = (const float*)d_in[0];
  const float* ea    = (const float*)d_in[1];
  const int*   eidx  = (const int*)d_in[2];
  const int*   batch = (const int*)d_in[3];
  const int*   srcI  = eidx;
  const int*   dstI  = eidx + EE;

  const float *Wl[3], *bl[3], *Wr[3], *br[3], *We[3], *att[3], *bo[3];
  for (int L = 0; L < 3; ++L) {
    int base = 4 + L * 7;
    Wl[L]  = (const float*)d_in[base + 0];
    bl[L]  = (const float*)d_in[base + 1];
    Wr[L]  = (const float*)d_in[base + 2];
    br[L]  = (const float*)d_in[base + 3];
    We[L]  = (const float*)d_in[base + 4];
    att[L] = (const float*)d_in[base + 5];
    bo[L]  = (const float*)d_in[base + 6];
  }
  const float* Wv1 = (const float*)d_in[25]; const float* bv1 = (const float*)d_in[26];
  const float* Wv2 = (const float*)d_in[27]; const float* bv2 = (const float*)d_in[28];
  const float* Wa1 = (const float*)d_in[29]; const float* ba1 = (const float*)d_in[30];
  const float* Wa2 = (const float*)d_in[31]; const float* ba2 = (const float*)d_in[32];

  // workspace layout (floats)
  float* ws     = (float*)d_ws;
  float* bufA   = ws;                         // N*128
  float* bufB   = bufA + (size_t)NN * FNODE;  // N*128
  float* xl     = bufB + (size_t)NN * FNODE;  // N*128
  float* xr     = xl   + (size_t)NN * FNODE;  // N*128
  float* score  = xr   + (size_t)NN * FNODE;  // E
  float* exv    = score + (size_t)EE;         // E
  float* mx     = exv  + (size_t)EE;          // N
  float* den    = mx   + (size_t)NN;          // N
  float* pooled = den  + (size_t)NN;          // G*128
  float* cnt    = pooled + (size_t)GG * FNODE;// G

  const int dinA[3]  = {FNODE, 32, 64};
  const int doutA[3] = {32, 64, 128};
  const int TB = 256;
  auto cdiv = [](int a, int b) { return (a + b - 1) / b; };

  const float* hin = x;
  for (int L = 0; L < 3; ++L) {
    int din = dinA[L], dout = doutA[L];
    float* agg = (L == 1) ? bufB : bufA;      // L0->A, L1->B, L2->A (ping-pong)

    // dense transforms via fp32 WMMA
    int ntilesN = dout / 16;
    int totalTiles = (NN / 16) * ntilesN;     // 50000/16 = 3125 exactly
    gemm_xlxr_wmma<<<cdiv(totalTiles, TB / 32), TB, 0, stream>>>(
        hin, Wl[L], bl[L], Wr[L], br[L], xl, xr, din, dout, ntilesN, totalTiles);

    // init reductions
    fill_f32<<<cdiv(NN, TB), TB, 0, stream>>>(mx, -INFINITY, NN);
    fill_f32<<<cdiv(NN, TB), TB, 0, stream>>>(den, 0.0f, NN);
    fill_f32<<<cdiv(NN * dout, TB), TB, 0, stream>>>(agg, 0.0f, NN * dout);

    // edge passes
    size_t shmem = (size_t)(FEDGE * dout + dout) * sizeof(float);
    edge_score<<<cdiv(EE, TB), TB, shmem, stream>>>(xl, xr, ea, srcI, dstI,
                                                    We[L], att[L], score, mx, dout);
    edge_exp<<<cdiv(EE, TB), TB, 0, stream>>>(score, mx, dstI, exv, den);
    edge_agg<<<cdiv(EE, TB), TB, 0, stream>>>(xl, exv, den, srcI, dstI, agg, dout);

    // out = relu(agg + bo)
    node_bias_relu<<<cdiv(NN * dout, TB), TB, 0, stream>>>(agg, bo[L], dout, NN * dout);
    hin = agg;
  }

  // global mean pool  (h3 lives in bufA)
  fill_f32<<<cdiv(GG * FNODE, TB), TB, 0, stream>>>(pooled, 0.0f, GG * FNODE);
  fill_f32<<<1, GG, 0, stream>>>(cnt, 0.0f, GG);
  pool_count<<<cdiv(NN, TB), TB, 0, stream>>>(batch, cnt);
  pool_sum<<<cdiv(NN * FNODE, TB), TB, 0, stream>>>(hin, batch, pooled);

  // dueling heads
  heads<<<GG, 128, 0, stream>>>(pooled, cnt, Wv1, bv1, Wv2, bv2,
                                Wa1, ba1, Wa2, ba2, (float*)d_out);
}